// CloudRasterizerOversample_71880572666287
// MI455X (gfx1250) — compile-verified
//
#include <hip/hip_runtime.h>

// Native clang vector type: accepted by __builtin_nontemporal_load
// (HIP_vector_type float4 is a struct and is rejected).
typedef float v4f __attribute__((ext_vector_type(4)));

// ---- grid constants (double math to mirror Python, then cast to f32) ----
#define N_PIX_HI 512
#define NV_HI 256
#define N_PIX_LO 128
#define NV_LO 64

__device__ __constant__ const float kPixHi     = (float)(0.1 / 4.0);                          // 0.025
__device__ __constant__ const float kFovHalfHi = (float)(0.5 * (512.0 - 1.0) * (0.1 / 4.0));  // 6.3875
__device__ __constant__ const float kDvHi      = (float)(12.5 / 4.0);                         // 3.125
__device__ __constant__ const float kVel0Hi    = (float)(-400.0 - 0.5 * (12.5 - 12.5 / 4.0)); // -404.6875

// Zero-init the 64*128*128 output (harness poisons d_out with 0xAA).
__global__ __launch_bounds__(256) void zero_out_kernel(v4f* __restrict__ out4, int n4) {
    int i = blockIdx.x * blockDim.x + threadIdx.x;
    if (i < n4) out4[i] = (v4f){0.f, 0.f, 0.f, 0.f};
}

// Per axis: decompose the two hi-res corners into 1 or 2 low-res (cell, weight)
// pairs. Both corners share a low-res cell unless (i0 & 3) == 3.
__device__ __forceinline__ int axis_cells(int i0, float f, int cell[2], float w[2]) {
    cell[0] = i0 >> 2;
    w[0]    = 1.0f;
    if ((i0 & 3) == 3) {
        w[0]    = 1.0f - f;
        cell[1] = cell[0] + 1;
        w[1]    = f;
        return 2;
    }
    return 1;
}

__device__ __forceinline__ void splat_point(float r, float d, float v, float fl,
                                            float* __restrict__ out) {
    // Exact same f32 ops as the reference (division, floorf) so bin
    // assignment matches bit-for-bit at cell boundaries.
    float x  = (r + kFovHalfHi) / kPixHi;
    float y  = (d + kFovHalfHi) / kPixHi;
    float z  = (v - kVel0Hi) / kDvHi;

    float xf = floorf(x), yf = floorf(y), zf = floorf(z);
    int ix0 = (int)xf, iy0 = (int)yf, iv0 = (int)zf;
    float fx = x - xf, fy = y - yf, fv = z - zf;

    // valid <=> i0 in [0, N-2]; invalid points contribute zero (skip).
    if ((unsigned)ix0 >= (unsigned)(N_PIX_HI - 1) ||
        (unsigned)iy0 >= (unsigned)(N_PIX_HI - 1) ||
        (unsigned)iv0 >= (unsigned)(NV_HI - 1))
        return;

    float fs = fl * (1.0f / 64.0f);

    int   cx[2], cy[2], cv[2];
    float wx[2], wy[2], wv[2];
    int nx = axis_cells(ix0, fx, cx, wx);
    int ny = axis_cells(iy0, fy, cy, wy);
    int nv = axis_cells(iv0, fv, cv, wv);

    // Expected ~1.95 native fp32 L2 atomics per point (vs 8 naive corners).
    for (int a = 0; a < nv; ++a) {
        float wa    = fs * wv[a];
        int   baseV = cv[a] * (N_PIX_LO * N_PIX_LO);
        for (int b = 0; b < ny; ++b) {
            float wab  = wa * wy[b];
            int   base = baseV + cy[b] * N_PIX_LO;
            for (int c = 0; c < nx; ++c) {
                // no-return global_atomic_add_f32 (not a CAS loop)
                unsafeAtomicAdd(out + base + cx[c], wab * wx[c]);
            }
        }
    }
}

// 4 points per thread: b128 non-temporal streaming loads (full cacheline
// utilization, 4x fewer load instructions), atomics into L2-resident output.
__global__ __launch_bounds__(256) void splat_kernel4(
    const v4f* __restrict__ ra4, const v4f* __restrict__ dec4,
    const v4f* __restrict__ vel4, const v4f* __restrict__ flux4,
    float* __restrict__ out, int n4)
{
    int i = blockIdx.x * blockDim.x + threadIdx.x;
    if (i >= n4) return;

    v4f r  = __builtin_nontemporal_load(ra4 + i);
    v4f d  = __builtin_nontemporal_load(dec4 + i);
    v4f v  = __builtin_nontemporal_load(vel4 + i);
    v4f fl = __builtin_nontemporal_load(flux4 + i);

    splat_point(r.x, d.x, v.x, fl.x, out);
    splat_point(r.y, d.y, v.y, fl.y, out);
    splat_point(r.z, d.z, v.z, fl.z, out);
    splat_point(r.w, d.w, v.w, fl.w, out);
}

// Scalar tail (M not divisible by 4).
__global__ __launch_bounds__(256) void splat_kernel_tail(
    const float* __restrict__ ra, const float* __restrict__ dec,
    const float* __restrict__ vel, const float* __restrict__ flux,
    float* __restrict__ out, int start, int M)
{
    int i = start + blockIdx.x * blockDim.x + threadIdx.x;
    if (i >= M) return;
    splat_point(ra[i], dec[i], vel[i], flux[i], out);
}

extern "C" void kernel_launch(void* const* d_in, const int* in_sizes, int n_in,
                              void* d_out, int out_size, void* d_ws, size_t ws_size,
                              hipStream_t stream) {
    const float* ra   = (const float*)d_in[0];
    const float* dec  = (const float*)d_in[1];
    const float* vel  = (const float*)d_in[2];
    const float* flux = (const float*)d_in[3];
    float* out = (float*)d_out;

    int M = in_sizes[0];

    // 1) zero the output (out_size = 64*128*128 = 1,048,576; multiple of 4)
    int nOut4 = out_size / 4;
    zero_out_kernel<<<(nOut4 + 255) / 256, 256, 0, stream>>>((v4f*)out, nOut4);

    // 2) merged-corner scatter, 4 points per thread
    int n4 = M / 4;
    if (n4 > 0) {
        splat_kernel4<<<(n4 + 255) / 256, 256, 0, stream>>>(
            (const v4f*)ra, (const v4f*)dec, (const v4f*)vel,
            (const v4f*)flux, out, n4);
    }
    int tail = M - n4 * 4;
    if (tail > 0) {
        splat_kernel_tail<<<1, 256, 0, stream>>>(ra, dec, vel, flux, out, n4 * 4, M);
    }
}